// AttentionModel_47313359733153
// MI455X (gfx1250) — compile-verified
//
#include <hip/hip_runtime.h>
#include <hip/hip_bf16.h>
#include <math.h>

// Shapes from the reference
#define BB   512
#define NN   101
#define NP   112      // N padded to multiple of 16
#define TT   64
#define EMB  128
#define HH   8
#define HD   16
#define CDIM 131      // step-context feature dim
#define NEGV (-1e9f)

typedef float v2f __attribute__((ext_vector_type(2)));
typedef float v8f __attribute__((ext_vector_type(8)));

// One f32 WMMA step: D = A(16x4) * B(4x16) + C, full-precision accumulation.
__device__ __forceinline__ v8f wmma4(v2f a, v2f b, v8f c) {
  return __builtin_amdgcn_wmma_f32_16x16x4_f32(
      /*neg_a=*/false, a, /*neg_b=*/false, b,
      /*c_mod=*/(short)0, c, /*reuse_a=*/false, /*reuse_b=*/false);
}

// ---------------------------------------------------------------------------
// Kernel 1: kvq[b, n(pad 112), 0:384] = node_embed[b,n,:] @ W_node  (K=128)
// One wave computes a 16x64 output strip (4 WMMA tiles) so each A fetch feeds
// 4 WMMAs. Pad rows (n >= 101) produce zeros (branchless select, loads safe).
// grid = (B*112/16, 384/64), block = 32
// ---------------------------------------------------------------------------
__global__ void k_kvq(const float* __restrict__ node,
                      const float* __restrict__ Wn,
                      float* __restrict__ kvq) {
  const int lane = threadIdx.x & 31, half = lane >> 4, l16 = lane & 15;
  const int m0 = blockIdx.x * 16, ng = blockIdx.y * 64;

  const int r = m0 + l16;               // padded row
  const int b = r / NP, n = r % NP;
  const bool av = (n < NN);
  const float* arow = node + ((size_t)b * NN + (av ? n : 0)) * EMB; // safe addr

  v8f acc[4] = {};
#pragma unroll
  for (int k0 = 0; k0 < EMB; k0 += 4) {
    const int ka = k0 + 2 * half;       // ISA f32 A layout: half picks K pair
    v2f a = *(const v2f*)(arow + ka);   // unconditional b64 load
    a.x = av ? a.x : 0.f;               // branchless pad-row zeroing
    a.y = av ? a.y : 0.f;
    const float* w0 = Wn + (size_t)ka * 384 + ng + l16;
#pragma unroll
    for (int j = 0; j < 4; ++j) {
      v2f w;
      w.x = w0[j * 16];
      w.y = w0[j * 16 + 384];
      acc[j] = wmma4(a, w, acc[j]);
    }
  }
#pragma unroll
  for (int j = 0; j < 4; ++j)
#pragma unroll
    for (int i = 0; i < 8; ++i) {
      const int rr = m0 + i + 8 * half; // C layout: VGPR i -> row i / i+8
      kvq[(size_t)rr * 384 + ng + j * 16 + l16] = acc[j][i];
    }
}

// ---------------------------------------------------------------------------
// Kernel 2: graph mean + fixed_context = mean(node) @ W_fixed  (tiny, VALU)
// grid = B, block = 128
// ---------------------------------------------------------------------------
__global__ void k_fixed(const float* __restrict__ node,
                        const float* __restrict__ Wf,
                        float* __restrict__ fixedc) {
  __shared__ float ge[EMB];
  const int b = blockIdx.x, e = threadIdx.x;
  float s = 0.f;
  for (int n = 0; n < NN; ++n) s += node[((size_t)b * NN + n) * EMB + e];
  ge[e] = s * (1.0f / (float)NN);
  __syncthreads();
  float acc = 0.f;
  for (int k = 0; k < EMB; ++k) acc += ge[k] * Wf[k * EMB + e];
  fixedc[b * EMB + e] = acc;
}

// ---------------------------------------------------------------------------
// Kernel 3: query[b,t,:] = fixed_context[b,:] + step_context[b,t,:] @ W_step
// Main K loop 0..127 unconditional; explicit tail step covers k=128..130 with
// clamped-safe addresses and a single branchless zero-select for k==131.
// grid = (B*T/16, 2) — each wave does a 16x64 strip. block = 32
// ---------------------------------------------------------------------------
__global__ void k_query(const float* __restrict__ step,
                        const float* __restrict__ Ws,
                        const float* __restrict__ fixedc,
                        float* __restrict__ query) {
  const int lane = threadIdx.x & 31, half = lane >> 4, l16 = lane & 15;
  const int m0 = blockIdx.x * 16, ng = blockIdx.y * 64;
  const float* arow = step + (size_t)(m0 + l16) * CDIM;

  v8f acc[4] = {};
#pragma unroll 8
  for (int k0 = 0; k0 < 128; k0 += 4) {
    const int ka = k0 + 2 * half;
    v2f a;                               // rows are 131 floats -> keep b32 loads
    a.x = arow[ka];
    a.y = arow[ka + 1];
    const float* w0 = Ws + (size_t)ka * EMB + ng + l16;
#pragma unroll
    for (int j = 0; j < 4; ++j) {
      v2f w;
      w.x = w0[j * 16];
      w.y = w0[j * 16 + EMB];
      acc[j] = wmma4(a, w, acc[j]);
    }
  }
  { // tail: k = 128..130 valid, k = 131 zero-padded
    const int ka = 128 + 2 * half;       // 128 (half 0) or 130 (half 1)
    v2f a;
    a.x = arow[ka];                      // 128 / 130: always valid
    const float ay = arow[ka + (half ? 0 : 1)]; // safe addr (129 or 130)
    a.y = half ? 0.f : ay;               // zero out k==131 product
    const int kb1 = half ? 130 : 129;    // safe W row; product zeroed by a.y
    const float* w0 = Ws + (size_t)ka * EMB + ng + l16;
    const float* w1 = Ws + (size_t)kb1 * EMB + ng + l16;
#pragma unroll
    for (int j = 0; j < 4; ++j) {
      v2f w;
      w.x = w0[j * 16];
      w.y = w1[j * 16];
      acc[j] = wmma4(a, w, acc[j]);
    }
  }
  const int b = m0 / TT;                 // tile lies inside one batch
#pragma unroll
  for (int j = 0; j < 4; ++j) {
    const float fc = fixedc[b * EMB + ng + j * 16 + l16];
#pragma unroll
    for (int i = 0; i < 8; ++i) {
      const int rr = m0 + i + 8 * half;
      query[(size_t)rr * EMB + ng + j * 16 + l16] = acc[j][i] + fc;
    }
  }
}

// ---------------------------------------------------------------------------
// Kernel 4: masked multi-head glimpse attention.
// grid = (B, T/16), block = 256 (8 waves, wave == head).
// Per wave: compat = Q_h(16x16) @ gK_h^T (7 node-tiles, WMMA),
// row softmax in registers via width-16 shuffles, attn staged in LDS,
// heads_h = attn(16x112) @ gV_h(112x16) via 28 WMMA steps.
// ---------------------------------------------------------------------------
__global__ void k_attn(const float* __restrict__ kvq,
                       const float* __restrict__ query,
                       const unsigned char* __restrict__ mask,
                       float* __restrict__ headsb) {
  __shared__ float attn_lds[HH][16][NP];   // 56 KB
  const int b = blockIdx.x, t0 = blockIdx.y * 16;
  const int h = threadIdx.x >> 5;
  const int lane = threadIdx.x & 31, half = lane >> 4, l16 = lane & 15;

  const float* gK = kvq + (size_t)b * NP * 384;        // cols [0,128)
  const float* gV = gK + EMB;                          // cols [128,256)
  const float* q  = query + ((size_t)b * TT + t0) * EMB + h * HD;

  // --- compat tiles: 7 x (16t x 16n), K = HD = 16 ---
  v8f ct[7];
#pragma unroll
  for (int nt = 0; nt < 7; ++nt) {
    v8f c = {};
#pragma unroll
    for (int k0 = 0; k0 < HD; k0 += 4) {
      const int ka = k0 + 2 * half;
      v2f a = *(const v2f*)(q + l16 * EMB + ka);
      // B[k][n] = gK[n0+n][h*16+k] (transposed read, 8B-aligned pair)
      v2f w = *(const v2f*)(gK + (size_t)(nt * 16 + l16) * 384 + h * HD + ka);
      c = wmma4(a, w, c);
    }
    ct[nt] = c;
  }

  // --- mask + scale + row softmax (rows live across 16 lanes of one half) ---
  const float inv_sqrt_d = 0.25f;                       // 1/sqrt(16)
#pragma unroll
  for (int i = 0; i < 8; ++i) {
    const int t = t0 + i + 8 * half;
    float m = NEGV;
#pragma unroll
    for (int nt = 0; nt < 7; ++nt) {
      const int n = nt * 16 + l16;
      const unsigned char mv = mask[((size_t)b * TT + t) * NN + (n < NN ? n : 0)];
      const bool mk = (n < NN) & (mv != 0);             // branchless
      const float v = mk ? ct[nt][i] * inv_sqrt_d : NEGV;
      ct[nt][i] = v;
      m = fmaxf(m, v);
    }
    for (int off = 8; off >= 1; off >>= 1) m = fmaxf(m, __shfl_xor(m, off, 16));
    float s = 0.f;
#pragma unroll
    for (int nt = 0; nt < 7; ++nt) {
      const float e = expf(ct[nt][i] - m);
      ct[nt][i] = e;
      s += e;
    }
    for (int off = 8; off >= 1; off >>= 1) s += __shfl_xor(s, off, 16);
    const float inv = 1.f / s;
#pragma unroll
    for (int nt = 0; nt < 7; ++nt)
      attn_lds[h][i + 8 * half][nt * 16 + l16] = ct[nt][i] * inv;
  }
  __syncthreads();

  // --- heads_h = attn(16 x 112) @ gV_h(112 x 16), K over nodes ---
  v8f hc = {};
#pragma unroll
  for (int k0 = 0; k0 < NP; k0 += 4) {
    const int ka = k0 + 2 * half;
    v2f a = *(const v2f*)(&attn_lds[h][l16][ka]);       // ds_load_b64
    v2f w;
    w.x = gV[(size_t)ka * 384 + h * HD + l16];
    w.y = gV[(size_t)(ka + 1) * 384 + h * HD + l16];
    hc = wmma4(a, w, hc);
  }
#pragma unroll
  for (int i = 0; i < 8; ++i) {
    const int t = t0 + i + 8 * half;
    headsb[((size_t)b * TT + t) * EMB + h * HD + l16] = hc[i];
  }
}

// ---------------------------------------------------------------------------
// Kernel 5: final_Q = heads @ W_out ; logits = final_Q @ logit_K^T /sqrt(E);
// tanh clip, mask, log_softmax over nodes.
// grid = (B, T/16), block = 256 (8 waves).
// ---------------------------------------------------------------------------
__global__ void k_out(const float* __restrict__ kvq,
                      const float* __restrict__ headsb,
                      const unsigned char* __restrict__ mask,
                      const float* __restrict__ Wout,
                      float* __restrict__ out) {
  __shared__ float fq[16][EMB];   // final_Q tile (8 KB)
  __shared__ float lg[16][NP];    // logits tile (7 KB)
  const int b = blockIdx.x, t0 = blockIdx.y * 16;
  const int wave = threadIdx.x >> 5;
  const int lane = threadIdx.x & 31, half = lane >> 4, l16 = lane & 15;
  const float* hrow = headsb + ((size_t)b * TT + t0) * EMB;

  // final_Q: wave w computes cols [16w, 16w+16), K = 128
  {
    const int n0 = wave * 16;
    v8f c = {};
#pragma unroll 8
    for (int k0 = 0; k0 < EMB; k0 += 4) {
      const int ka = k0 + 2 * half;
      v2f a = *(const v2f*)(hrow + l16 * EMB + ka);
      v2f w;
      w.x = Wout[(size_t)ka * EMB + n0 + l16];
      w.y = Wout[(size_t)(ka + 1) * EMB + n0 + l16];
      c = wmma4(a, w, c);
    }
#pragma unroll
    for (int i = 0; i < 8; ++i) fq[i + 8 * half][n0 + l16] = c[i];
  }
  __syncthreads();

  // logits: waves 0..6 each take one node-tile (wave-uniform branch, EXEC full)
  if (wave < 7) {
    const int n0 = wave * 16;
    const float* lK = kvq + (size_t)b * NP * 384 + 256;   // logit_K cols
    v8f c = {};
#pragma unroll 8
    for (int k0 = 0; k0 < EMB; k0 += 4) {
      const int ka = k0 + 2 * half;
      v2f a = *(const v2f*)(&fq[l16][ka]);
      v2f w = *(const v2f*)(lK + (size_t)(n0 + l16) * 384 + ka); // B[k][n]=logit_K[n][k]
      c = wmma4(a, w, c);
    }
    const float inv_sqrt_e = 0.08838834764831845f;        // 1/sqrt(128)
#pragma unroll
    for (int i = 0; i < 8; ++i) {
      const int t = t0 + i + 8 * half, n = n0 + l16;
      const float v = 10.f * tanhf(c[i] * inv_sqrt_e);
      const unsigned char mv = mask[((size_t)b * TT + t) * NN + (n < NN ? n : 0)];
      const bool mk = (n < NN) & (mv != 0);
      lg[i + 8 * half][n] = mk ? v : NEGV;
    }
  }
  __syncthreads();

  // log_softmax: 16 teams of 16 lanes, team r owns row r
  const int team = threadIdx.x >> 4, tl = threadIdx.x & 15;
  float m = NEGV;
  for (int n = tl; n < NN; n += 16) m = fmaxf(m, lg[team][n]);
  for (int off = 8; off >= 1; off >>= 1) m = fmaxf(m, __shfl_xor(m, off, 16));
  float s = 0.f;
  for (int n = tl; n < NN; n += 16) s += expf(lg[team][n] - m);
  for (int off = 8; off >= 1; off >>= 1) s += __shfl_xor(s, off, 16);
  const float lse = m + logf(s);
  for (int n = tl; n < NN; n += 16)
    out[((size_t)b * TT + t0 + team) * NN + n] = lg[team][n] - lse;
}

// ---------------------------------------------------------------------------
extern "C" void kernel_launch(void* const* d_in, const int* in_sizes, int n_in,
                              void* d_out, int out_size, void* d_ws, size_t ws_size,
                              hipStream_t stream) {
  const float* node = (const float*)d_in[0];                 // [512,101,128]
  const float* step = (const float*)d_in[1];                 // [512,64,131]
  const unsigned char* mask = (const unsigned char*)d_in[2]; // [512,64,101] bool
  const float* Wn = (const float*)d_in[3];                   // [128,384]
  const float* Wf = (const float*)d_in[4];                   // [128,128]
  const float* Ws = (const float*)d_in[5];                   // [131,128]
  const float* Wo = (const float*)d_in[6];                   // [128,128]
  float* out = (float*)d_out;

  // workspace layout (floats)
  float* ws     = (float*)d_ws;
  float* kvq    = ws;                                        // 512*112*384
  float* fixedc = kvq + (size_t)BB * NP * 384;               // 512*128
  float* query  = fixedc + (size_t)BB * EMB;                 // 512*64*128
  float* headsb = query + (size_t)BB * TT * EMB;             // 512*64*128

  k_kvq  <<<dim3((BB * NP) / 16, 384 / 64), 32,  0, stream>>>(node, Wn, kvq);
  k_fixed<<<BB, EMB, 0, stream>>>(node, Wf, fixedc);
  k_query<<<dim3((BB * TT) / 16, 2), 32, 0, stream>>>(step, Ws, fixedc, query);
  k_attn <<<dim3(BB, TT / 16), 256, 0, stream>>>(kvq, query, mask, headsb);
  k_out  <<<dim3(BB, TT / 16), 256, 0, stream>>>(kvq, headsb, mask, Wo, out);
}